// GATLayer_84335977824817
// MI455X (gfx1250) — compile-verified
//
#include <hip/hip_runtime.h>
#include <math.h>

typedef __attribute__((ext_vector_type(2))) float v2f;
typedef __attribute__((ext_vector_type(8))) float v8f;

#define N_NODES 50000
#define N_EDGES 800000
#define IN_F    128
#define OUT_F   32
#define HEADS   4
#define EDGE_F  32
#define HF      (HEADS * OUT_F)   // 128
#define A_ROW   (2 * OUT_F + EDGE_F) // 96

// ---- order-preserving float<->uint encoding for atomic max over floats ----
__device__ __forceinline__ unsigned enc_f32(float f) {
  unsigned u = __float_as_uint(f);
  return (u & 0x80000000u) ? ~u : (u | 0x80000000u);
}
__device__ __forceinline__ float dec_f32(unsigned u) {
  unsigned b = (u & 0x80000000u) ? (u ^ 0x80000000u) : ~u;
  return __uint_as_float(b);
}

// ---- 0: zero accumulators, seed global max (re-run every call: graph-safe) ----
__global__ void k_init(float* __restrict__ hout, float* __restrict__ denom,
                       unsigned* __restrict__ gmax) {
  int idx = blockIdx.x * blockDim.x + threadIdx.x;
  if (idx < N_NODES * HF) hout[idx] = 0.f;
  if (idx < N_NODES * HEADS) denom[idx] = 0.f;
  if (idx == 0) *gmax = enc_f32(-3.402823466e38f);
}

// ---- 1: ht = h @ W via V_WMMA_F32_16X16X4_F32 ----
// Block = 256 threads = 8 waves. Block b computes rows [16b,16b+16), wave w
// computes columns [16w,16w+16). K-loop: 128/4 = 32 WMMA ops, fp32 accum.
__global__ void __launch_bounds__(256) k_gemm_ht(const float* __restrict__ h,
                                                 const float* __restrict__ W,
                                                 float* __restrict__ ht) {
  const int lane = threadIdx.x & 31;
  const int wave = threadIdx.x >> 5;
  const int m0 = blockIdx.x * 16;
  const int n0 = wave * 16;
  const int kp = (lane >> 4) * 2;   // lanes 0-15 hold K={0,1}, lanes 16-31 K={2,3}
  const int mr = lane & 15;

  v8f c = {};
  const float* arow = h + (size_t)(m0 + mr) * IN_F + kp;     // A: 16x4 tile
  const float* bcol = W + (size_t)kp * HF + n0 + mr;          // B: 4x16 tile

  for (int k0 = 0; k0 < IN_F; k0 += 4) {
    v2f A;
    A.x = arow[k0];
    A.y = arow[k0 + 1];
    v2f B;
    B.x = bcol[(size_t)k0 * HF];
    B.y = bcol[(size_t)(k0 + 1) * HF];
    c = __builtin_amdgcn_wmma_f32_16x16x4_f32(false, A, false, B,
                                              (short)0, c, false, false);
  }

  // C/D layout: VGPR r -> M = r + 8*(lane>=16), N = lane&15
  const int rbase = m0 + ((lane >> 4) * 8);
  const int col = n0 + mr;
#pragma unroll
  for (int r = 0; r < 8; ++r)
    ht[(size_t)(rbase + r) * HF + col] = c[r];
}

// ---- 2: fold eW/eb through a_e:  v[h][k], b[h] ----
__global__ void k_prep(const float* __restrict__ a, const float* __restrict__ eW,
                       const float* __restrict__ eb, float* __restrict__ v,
                       float* __restrict__ bvec) {
  int t = blockIdx.x * blockDim.x + threadIdx.x;
  if (t < HEADS * EDGE_F) {
    int hh = t / EDGE_F, k = t % EDGE_F;
    const float* ae = a + hh * A_ROW + 2 * OUT_F;
    float s = 0.f;
    for (int f = 0; f < OUT_F; ++f) s += eW[k * HF + hh * OUT_F + f] * ae[f];
    v[t] = s;
  }
  if (t < HEADS) {
    const float* ae = a + t * A_ROW + 2 * OUT_F;
    float s = 0.f;
    for (int f = 0; f < OUT_F; ++f) s += eb[t * OUT_F + f] * ae[f];
    bvec[t] = s;
  }
}

// ---- 3: alpha_i / alpha_j per (node, head) ----
__global__ void k_alpha(const float* __restrict__ ht, const float* __restrict__ a,
                        float* __restrict__ ai, float* __restrict__ aj) {
  int t = blockIdx.x * blockDim.x + threadIdx.x;
  if (t >= N_NODES * HEADS) return;
  int n = t >> 2, hh = t & 3;
  const float* rowp = ht + (size_t)n * HF + hh * OUT_F;
  const float* aih = a + hh * A_ROW;
  const float* ajh = aih + OUT_F;
  float si = 0.f, sj = 0.f;
  for (int f = 0; f < OUT_F; ++f) {
    float x = rowp[f];
    si += x * aih[f];
    sj += x * ajh[f];
  }
  ai[t] = si;
  aj[t] = sj;
}

// ---- 4: per-edge raw scores + leaky relu + global max ----
__global__ void __launch_bounds__(256) k_escore(
    const int* __restrict__ row, const int* __restrict__ col,
    const float* __restrict__ edge_attr, const float* __restrict__ ai,
    const float* __restrict__ aj, const float* __restrict__ v,
    const float* __restrict__ bvec, float* __restrict__ esc,
    unsigned* __restrict__ gmax) {
  __shared__ float smax[256];
  int e = blockIdx.x * blockDim.x + threadIdx.x;
  float lmax = -3.402823466e38f;
  if (e < N_EDGES) {
    int r = row[e], cN = col[e];
    float s[HEADS];
#pragma unroll
    for (int hh = 0; hh < HEADS; ++hh)
      s[hh] = ai[r * HEADS + hh] + aj[cN * HEADS + hh] + bvec[hh];
    const float* ea = edge_attr + (size_t)e * EDGE_F;
    for (int k = 0; k < EDGE_F; ++k) {
      float x = ea[k];
#pragma unroll
      for (int hh = 0; hh < HEADS; ++hh) s[hh] += x * v[hh * EDGE_F + k];
    }
#pragma unroll
    for (int hh = 0; hh < HEADS; ++hh) {
      float val = s[hh] > 0.f ? s[hh] : 0.2f * s[hh];  // leaky_relu 0.2
      esc[e * HEADS + hh] = val;
      lmax = fmaxf(lmax, val);
    }
  }
  smax[threadIdx.x] = lmax;
  __syncthreads();
  for (int off = 128; off > 0; off >>= 1) {
    if (threadIdx.x < off)
      smax[threadIdx.x] = fmaxf(smax[threadIdx.x], smax[threadIdx.x + off]);
    __syncthreads();
  }
  if (threadIdx.x == 0) atomicMax(gmax, enc_f32(smax[0]));
}

// ---- 5: exp(score - max) + segment-sum denominators ----
__global__ void k_eexp(const int* __restrict__ row, float* __restrict__ esc,
                       float* __restrict__ denom, const unsigned* __restrict__ gmax) {
  int t = blockIdx.x * blockDim.x + threadIdx.x;
  if (t >= N_EDGES * HEADS) return;
  float m = dec_f32(*gmax);
  int e = t >> 2, hh = t & 3;
  float val = __expf(esc[t] - m);
  esc[t] = val;
  atomicAdd(&denom[row[e] * HEADS + hh], val);
}

// ---- 6: att output + scatter att*ht[col] into h_out (32 lanes per edge) ----
__global__ void __launch_bounds__(256) k_scatter(
    const int* __restrict__ row, const int* __restrict__ col,
    const float* __restrict__ esc, const float* __restrict__ denom,
    const float* __restrict__ ht, float* __restrict__ hout,
    float* __restrict__ att_out) {
  long long t = (long long)blockIdx.x * blockDim.x + threadIdx.x;
  if (t >= (long long)N_EDGES * 32) return;
  int e = (int)(t >> 5);
  int i = (int)(t & 31);       // covers 4 floats each: f = 4i .. 4i+3
  int hh = i >> 3;             // head = (4i)/32
  int r = row[e], cN = col[e];
  float att = esc[e * HEADS + hh] / (denom[r * HEADS + hh] + 1e-8f);
  if ((i & 7) == 0) att_out[e * HEADS + hh] = att;
  const float4 hv = *(const float4*)(ht + (size_t)cN * HF + i * 4);
  float* dst = hout + (size_t)r * HF + i * 4;
  atomicAdd(dst + 0, att * hv.x);
  atomicAdd(dst + 1, att * hv.y);
  atomicAdd(dst + 2, att * hv.z);
  atomicAdd(dst + 3, att * hv.w);
}

// ---- 7: ELU in place on h_out ----
__global__ void k_elu(float* __restrict__ hout) {
  int t = blockIdx.x * blockDim.x + threadIdx.x;
  if (t >= N_NODES * HF) return;
  float x = hout[t];
  hout[t] = x > 0.f ? x : (__expf(x) - 1.f);
}

extern "C" void kernel_launch(void* const* d_in, const int* in_sizes, int n_in,
                              void* d_out, int out_size, void* d_ws, size_t ws_size,
                              hipStream_t stream) {
  (void)in_sizes; (void)n_in; (void)out_size; (void)ws_size;
  const float* h  = (const float*)d_in[0];
  const int*   ei = (const int*)d_in[1];
  const float* ea = (const float*)d_in[2];
  const float* W  = (const float*)d_in[3];
  const float* a  = (const float*)d_in[4];
  const float* eW = (const float*)d_in[5];
  const float* eb = (const float*)d_in[6];
  const int* row = ei;
  const int* col = ei + N_EDGES;

  float* hout = (float*)d_out;                                 // N*128
  float* att  = (float*)d_out + (size_t)N_NODES * HF;          // E*4

  char* ws = (char*)d_ws;
  float* ht  = (float*)ws; ws += (size_t)N_NODES * HF * 4;     // 25.6 MB
  float* ai  = (float*)ws; ws += (size_t)N_NODES * HEADS * 4;
  float* aj  = (float*)ws; ws += (size_t)N_NODES * HEADS * 4;
  float* esc = (float*)ws; ws += (size_t)N_EDGES * HEADS * 4;  // 12.8 MB
  float* den = (float*)ws; ws += (size_t)N_NODES * HEADS * 4;
  float* v   = (float*)ws; ws += HEADS * EDGE_F * 4;
  float* bv  = (float*)ws; ws += HEADS * 4;
  unsigned* gmax = (unsigned*)ws;

  k_init<<<(N_NODES * HF + 255) / 256, 256, 0, stream>>>(hout, den, gmax);
  k_gemm_ht<<<N_NODES / 16, 256, 0, stream>>>(h, W, ht);
  k_prep<<<1, 256, 0, stream>>>(a, eW, eb, v, bv);
  k_alpha<<<(N_NODES * HEADS + 255) / 256, 256, 0, stream>>>(ht, a, ai, aj);
  k_escore<<<(N_EDGES + 255) / 256, 256, 0, stream>>>(row, col, ea, ai, aj, v, bv, esc, gmax);
  k_eexp<<<(N_EDGES * HEADS + 255) / 256, 256, 0, stream>>>(row, esc, den, gmax);
  long long tot = (long long)N_EDGES * 32;
  k_scatter<<<(unsigned)((tot + 255) / 256), 256, 0, stream>>>(row, col, esc, den, ht, hout, att);
  k_elu<<<(N_NODES * HF + 255) / 256, 256, 0, stream>>>(hout);
}